// DiffRasterizer_60885456388523
// MI455X (gfx1250) — compile-verified
//
#include <hip/hip_runtime.h>
#include <hip/hip_bf16.h>
#include <math.h>

typedef __attribute__((ext_vector_type(2))) float v2f;
typedef __attribute__((ext_vector_type(8))) float v8f;

#define NSHAPE 64
#define NSAMP  30
#define SOFT   0.01f

// workspace float layout (all written every call, sorted by z ascending)
#define WS_POLY 0          // 64*30*2 = 3840 floats : [sortedPos][sample][dim]
#define WS_GATE 3840       // 64 floats
#define WS_COL  3904       // 192 floats
#define WS_CSG  4096       // 64 ints (reinterpret)

// ---------------------------------------------------------------------------
// Kernel A: z-argsort (stable O(N^2) rank), per-shape gates/colors/csg in
// sorted order, and bezier->polyline via WMMA:
//   D(32x128) = Bmat(32x12, rows>=30 zero) x Pmat(12x128)
// where column c -> (sortedShape = c>>1, dim = c&1). One block, 512 threads
// = 16 waves; wave w handles tile (mt = w>>3, nt = w&7).
// ---------------------------------------------------------------------------
__global__ __launch_bounds__(512)
void DiffRast_prep(const float* __restrict__ P, const float* __restrict__ c,
                   const float* __restrict__ alpha, const float* __restrict__ alive,
                   const float* __restrict__ z, const int* __restrict__ csg,
                   float* __restrict__ ws)
{
    __shared__ float sB[32][12];
    __shared__ int   sOrder[64];
    __shared__ float sZ[64];

    const int tid = threadIdx.x;
    if (tid < 64) sZ[tid] = z[tid];

    // Basis matrix: Bmat[s][j] = sum_k basis[s][k] * (idx[s][k]==j)
    if (tid < 384) {
        int s = tid / 12, j = tid % 12;
        float v = 0.f;
        if (s < NSAMP) {
            float tg = (float)s * (4.0f / 30.0f);          // linspace step == 4/30
            int seg = (int)floorf(tg);
            seg = seg < 0 ? 0 : (seg > 3 ? 3 : seg);
            float t = tg - (float)seg, ti = 1.f - t;
            float w0 = ti*ti*ti, w1 = 3.f*ti*ti*t, w2 = 3.f*ti*t*t, w3 = t*t*t;
            int i0 = seg*3, i3 = (i0 + 3) % 12;
            if (j == i0)     v += w0;
            if (j == i0 + 1) v += w1;
            if (j == i0 + 2) v += w2;
            if (j == i3)     v += w3;
        }
        sB[s][j] = v;
    }
    __syncthreads();

    // stable argsort of z (ascending)
    if (tid < 64) {
        float zi = sZ[tid];
        int r = 0;
        for (int j = 0; j < 64; ++j) {
            float zj = sZ[j];
            r += (zj < zi) || (zj == zi && j < tid);
        }
        sOrder[r] = tid;
    }
    __syncthreads();

    if (tid < 64) {
        int shp = sOrder[tid];
        // sigmoid(alive) > 0.1  <=>  alive > logit(0.1)
        float g = alpha[shp] * ((alive[shp] > -2.1972246f) ? 1.f : 0.f);
        ws[WS_GATE + tid]      = g;
        ws[WS_COL + tid*3 + 0] = c[shp*3 + 0];
        ws[WS_COL + tid*3 + 1] = c[shp*3 + 1];
        ws[WS_COL + tid*3 + 2] = c[shp*3 + 2];
        ((int*)ws)[WS_CSG + tid] = csg[shp];
    }

    // ---- WMMA polyline evaluation (uniform control flow, EXEC all ones) ----
    const int lane  = tid & 31;
    const int wv    = tid >> 5;
    const int mt    = wv >> 3;            // 0..1  (sample row tile)
    const int nt    = wv & 7;             // 0..7  (column tile)
    const int mrow  = lane & 15;
    const int khalf = (lane < 16) ? 0 : 2;
    const int cg    = nt*16 + (lane & 15);  // global column 0..127
    const int sp    = cg >> 1;              // sorted shape position
    const int dd    = cg & 1;               // dim (x/y)
    const int shp   = sOrder[sp];

    v8f acc = {};
#pragma unroll
    for (int k0 = 0; k0 < 12; k0 += 4) {
        v2f a, b;
        // A 16x4 f32: VGPR0 -> K = k0+khalf, VGPR1 -> K = k0+khalf+1 (M = mrow)
        a.x = sB[mt*16 + mrow][k0 + khalf];
        a.y = sB[mt*16 + mrow][k0 + khalf + 1];
        // B 4x16 f32: same K split across lane halves, N = lane&15
        b.x = P[shp*24 + (k0 + khalf)*2 + dd];
        b.y = P[shp*24 + (k0 + khalf + 1)*2 + dd];
        acc = __builtin_amdgcn_wmma_f32_16x16x4_f32(
            /*neg_a=*/false, a, /*neg_b=*/false, b,
            /*c_mod=*/(short)0, acc, /*reuse_a=*/false, /*reuse_b=*/false);
    }
    // D 16x16 f32: VGPR r -> M = r (lanes 0-15) / 8+r (lanes 16-31), N = lane&15
#pragma unroll
    for (int r = 0; r < 8; ++r) {
        int m = mt*16 + ((lane < 16) ? r : r + 8);
        if (m < NSAMP)
            ws[WS_POLY + sp*(NSAMP*2) + m*2 + dd] = acc[r];
    }
}

// ---------------------------------------------------------------------------
// Kernel B: one thread per pixel. Sorted polylines + per-shape scalars staged
// in LDS (~16.6 KB). Fused SDF + coverage + over/CSG composite per shape.
// ---------------------------------------------------------------------------
__global__ __launch_bounds__(256)
void DiffRast_render(const float* __restrict__ ws, float* __restrict__ out,
                     int W, int H)
{
    __shared__ float sPoly[NSHAPE * NSAMP * 2];
    __shared__ float sGate[NSHAPE];
    __shared__ float sCol[NSHAPE * 3];
    __shared__ int   sCsg[NSHAPE];

    const int tid = threadIdx.y * 16 + threadIdx.x;
    for (int i = tid; i < NSHAPE * NSAMP * 2; i += 256) sPoly[i] = ws[WS_POLY + i];
    if (tid < 64) {
        sGate[tid] = ws[WS_GATE + tid];
        sCsg[tid]  = ((const int*)ws)[WS_CSG + tid];
    }
    if (tid < 192) sCol[tid] = ws[WS_COL + tid];
    __syncthreads();

    const int x = blockIdx.x * 16 + threadIdx.x;
    const int y = blockIdx.y * 16 + threadIdx.y;
    if (x >= W || y >= H) return;

    const float px = (float)x / (float)(W - 1);
    const float py = (float)y / (float)(H - 1);

    float r = 0.f, g = 0.f, b = 0.f, ca = 0.f;

    for (int spi = 0; spi < NSHAPE; ++spi) {
        const float* pl = &sPoly[spi * NSAMP * 2];
        float mind = __builtin_inff();
        int wn = 0;
        float ax = pl[0], ay = pl[1];
#pragma unroll 5
        for (int s = 0; s < NSAMP; ++s) {
            int sn = (s + 1 == NSAMP) ? 0 : s + 1;
            float bx = pl[sn*2], by = pl[sn*2 + 1];
            float abx = bx - ax, aby = by - ay;
            float apx = px - ax, apy = py - ay;
            float denom = abx*abx + aby*aby + 1e-8f;
            float t = (apx*abx + apy*aby) / denom;
            t = t < 0.f ? 0.f : (t > 1.f ? 1.f : t);
            float dx = apx - t*abx, dy = apy - t*aby;
            float d = dx*dx + dy*dy;
            mind = fminf(mind, d);
            bool y1 = (ay <= py), y2 = (py < by);
            bool left = (abx*apy - apx*aby) > 0.f;   // cross((b-a),(p-a))
            wn += (y1 && y2 && left) ? 1 : 0;
            wn -= (!y1 && !y2 && !left) ? 1 : 0;
            ax = bx; ay = by;
        }
        float dist = sqrtf(mind + 1e-8f);
        float sdf  = (wn != 0) ? -dist : dist;
        float cov  = 1.f / (1.f + expf(sdf * (1.f / SOFT)));
        float la   = cov * sGate[spi];

        if (sCsg[spi]) {
            float can  = ca * (1.f - la);
            float keep = (can > 0.f) ? 1.f : 0.f;
            r *= keep; g *= keep; b *= keep;
            ca = can;
        } else {
            float oa   = la + ca * (1.f - la);
            float safe = (oa > 0.f) ? oa : 1.f;
            float w1   = la / safe;
            float w2   = ca * (1.f - la) / safe;
            float nr = sCol[spi*3 + 0] * w1 + r * w2;
            float ng = sCol[spi*3 + 1] * w1 + g * w2;
            float nb = sCol[spi*3 + 2] * w1 + b * w2;
            bool pos = oa > 0.f;
            r = pos ? nr : 0.f; g = pos ? ng : 0.f; b = pos ? nb : 0.f;
            ca = oa;
        }
    }

    int o = (y * W + x) * 3;
    out[o + 0] = fminf(fmaxf(r * ca, 0.f), 1.f);
    out[o + 1] = fminf(fmaxf(g * ca, 0.f), 1.f);
    out[o + 2] = fminf(fmaxf(b * ca, 0.f), 1.f);
}

extern "C" void kernel_launch(void* const* d_in, const int* in_sizes, int n_in,
                              void* d_out, int out_size, void* d_ws, size_t ws_size,
                              hipStream_t stream) {
    const float* P     = (const float*)d_in[0];
    const float* c     = (const float*)d_in[1];
    const float* alpha = (const float*)d_in[2];
    const float* alive = (const float*)d_in[3];
    const float* z     = (const float*)d_in[4];
    const int*   csg   = (const int*)d_in[5];
    (void)in_sizes; (void)n_in; (void)out_size; (void)ws_size;

    const int W = 512, H = 512;   // fixed by setup_inputs
    float* ws = (float*)d_ws;

    DiffRast_prep<<<1, 512, 0, stream>>>(P, c, alpha, alive, z, csg, ws);

    dim3 blk(16, 16);
    dim3 grd((W + 15) / 16, (H + 15) / 16);
    DiffRast_render<<<grd, blk, 0, stream>>>(ws, (float*)d_out, W, H);
}